// LatentReverb_335007450008
// MI455X (gfx1250) — compile-verified
//
#include <hip/hip_runtime.h>

typedef __attribute__((ext_vector_type(2))) float v2f;
typedef __attribute__((ext_vector_type(8))) float v8f;

// Problem constants (B=2, C=4, H=W=64, S=HW=4096, NH=4, hd=1)
#define HW_ 4096

__device__ __forceinline__ float sigmoidf_(float v) { return 1.0f / (1.0f + __expf(-v)); }

// fade_vec(64, fw=4): ramp = linspace(0.6,1.0,4), mirrored at the end
__device__ __forceinline__ float fade64(int i) {
    const float step = 0.4f / 3.0f;
    if (i < 4)   return 0.6f + step * (float)i;
    if (i >= 60) return 0.6f + step * (float)(63 - i);
    return 1.0f;
}

// CDNA5 async global->LDS copy (VGLOBAL GLOBAL_LOAD_ASYNC_TO_LDS_B32, ASYNCcnt-tracked).
// Per ISA 10.2 the low 32 bits of a generic LDS pointer are the LDS byte address, which is
// exactly what the VDST operand of the async load consumes.
__device__ __forceinline__ void async_load_f32_to_lds(const float* gaddr, float* laddr) {
    unsigned lds = (unsigned)(uintptr_t)laddr;
    asm volatile("global_load_async_to_lds_b32 %0, %1, off"
                 :: "v"(lds), "v"(gaddr) : "memory");
}
__device__ __forceinline__ void wait_asynccnt0() {
    asm volatile("s_wait_asynccnt 0x0" ::: "memory");
}

// ---------------------------------------------------------------- init
// fb0 = x * 0.1 ; wet = 0       (32768 threads)
__global__ void init_kernel(const float* __restrict__ x, float* __restrict__ fb0,
                            float* __restrict__ wet) {
    int i = blockIdx.x * blockDim.x + threadIdx.x;
    fb0[i] = x[i] * 0.1f;
    wet[i] = 0.0f;
}

// ---------------------------------------------------------------- spatial conv 3x3, 4->8, SAME
__global__ void spatial_conv_kernel(const float* __restrict__ x, const float* __restrict__ sp_w,
                                    const float* __restrict__ sp_b, float* __restrict__ spf) {
    __shared__ float s_w[288];
    __shared__ float s_b[8];
    int t = threadIdx.x;
    if (t < 288) s_w[t] = sp_w[t];
    if (t < 8)   s_b[t] = sp_b[t];
    __syncthreads();
    int tid = blockIdx.x * blockDim.x + t;      // 65536 = B*8*HW
    int b  = tid >> 15;
    int oc = (tid >> 12) & 7;
    int p  = tid & 4095;
    int y = p >> 6, x0 = p & 63;
    float acc = s_b[oc];
    #pragma unroll
    for (int ky = 0; ky < 3; ++ky) {
        int yy = y + ky - 1;
        if ((unsigned)yy >= 64u) continue;
        #pragma unroll
        for (int kx = 0; kx < 3; ++kx) {
            int xx = x0 + kx - 1;
            if ((unsigned)xx >= 64u) continue;
            #pragma unroll
            for (int ic = 0; ic < 4; ++ic) {
                float v = x[((b * 4 + ic) << 12) + (yy << 6) + xx];
                acc += v * s_w[((oc * 4 + ic) * 3 + ky) * 3 + kx];
            }
        }
    }
    spf[tid] = acc;
}

// ---------------------------------------------------------------- one reflection iteration (fused)
// refl = conv3x3(roll(fb_in)*fade) ; damping = sigmoid(d2(silu(d1(refl)))) ;
// wet += refl*damping*w*3 ; fb_out = fb_in + refl*damping*0.32
__global__ void reflection_step_kernel(const float* __restrict__ fb_in, float* __restrict__ fb_out,
                                       float* __restrict__ wet,
                                       const float* __restrict__ fb_w, const float* __restrict__ fb_b,
                                       const float* __restrict__ d1_w, const float* __restrict__ d1_b,
                                       const float* __restrict__ d2_w, const float* __restrict__ d2_b,
                                       const float* __restrict__ rw, const float* __restrict__ rd,
                                       int iter, float decay) {
    __shared__ float s_fw[144], s_fb[4], s_d1w[8], s_d1b[2], s_d2w[8], s_d2b[4];
    int t = threadIdx.x;
    if (t < 144) s_fw[t] = fb_w[t];
    if (t < 4)   s_fb[t] = fb_b[t];
    if (t < 8)   s_d1w[t] = d1_w[t];
    if (t < 2)   s_d1b[t] = d1_b[t];
    if (t >= 8  && t < 16) s_d2w[t - 8] = d2_w[t - 8];
    if (t >= 16 && t < 20) s_d2b[t - 16] = d2_b[t - 16];
    __syncthreads();

    int tid = blockIdx.x * blockDim.x + t;      // 8192 = B*HW
    int b = tid >> 12;
    int p = tid & 4095;
    int y = p >> 6, x0 = p & 63;

    float delay  = rd[iter];
    float scaled = delay * 4.0f;                // delay * ROOM_SIZE * 8
    int ys = (int)scaled;                       // <= 60
    int xs = (int)(scaled * 2.0f);              // <= 120
    float weight = sigmoidf_(rw[iter]) * decay * 1.2f;   // *(1+FEEDBACK*0.5)
    float wscale = weight * 3.0f;

    float refl[4] = { s_fb[0], s_fb[1], s_fb[2], s_fb[3] };
    #pragma unroll
    for (int ky = 0; ky < 3; ++ky) {
        int yy = y + ky - 1;
        if ((unsigned)yy >= 64u) continue;      // SAME zero pad
        float yf = fade64(yy);
        int ysrc = (yy - ys + 64) & 63;         // roll on read
        #pragma unroll
        for (int kx = 0; kx < 3; ++kx) {
            int xx = x0 + kx - 1;
            if ((unsigned)xx >= 64u) continue;
            float f = yf * fade64(xx);
            int xsrc = (xx - xs + 256) & 63;
            int base = (b << 14) + (ysrc << 6) + xsrc;
            #pragma unroll
            for (int ic = 0; ic < 4; ++ic) {
                float sv = fb_in[base + (ic << 12)] * f;
                #pragma unroll
                for (int oc = 0; oc < 4; ++oc)
                    refl[oc] += sv * s_fw[((oc * 4 + ic) * 3 + ky) * 3 + kx];
            }
        }
    }
    // damping MLP (1x1 convs: 4->2 silu, 2->4 sigmoid)
    float h0 = s_d1b[0], h1 = s_d1b[1];
    #pragma unroll
    for (int c = 0; c < 4; ++c) { h0 += refl[c] * s_d1w[c]; h1 += refl[c] * s_d1w[4 + c]; }
    h0 = h0 * sigmoidf_(h0);
    h1 = h1 * sigmoidf_(h1);
    #pragma unroll
    for (int oc = 0; oc < 4; ++oc) {
        float damp = sigmoidf_(s_d2b[oc] + h0 * s_d2w[oc * 2] + h1 * s_d2w[oc * 2 + 1]);
        float r = refl[oc] * damp;
        int idx = ((b * 4 + oc) << 12) + p;
        wet[idx] += r * wscale;
        fb_out[idx] = fb_in[idx] + r * 0.32f;   // FEEDBACK*0.8
    }
}

// ---------------------------------------------------------------- QKV projection via WMMA
// qkv[b,s,n] = sum_c wet[b,c,s]*in_w[n,c] + in_b[n]  (M=8192, K=4, N=12) -> v_wmma_f32_16x16x4_f32
// A layout (ISA 7.12.2): lane = M (0..15), VGPR0/1 = K; lanes 16..31 hold K=2,3.
// B layout: lane = N, VGPR0/1 = K (lanes 16..31 hold K=2,3).
// D layout: VGPR r -> M = r + 8*(lane>>4), N = lane&15.
__global__ void qkv_proj_wmma_kernel(const float* __restrict__ wet,
                                     const float* __restrict__ in_w, const float* __restrict__ in_b,
                                     float* __restrict__ qkv) {
    int lane = threadIdx.x & 31;
    int wid  = threadIdx.x >> 5;
    int tile = blockIdx.x * 8 + wid;            // 512 tiles of 16 rows
    int m0 = tile << 4;
    int b  = m0 >> 12;
    int s0 = m0 & 4095;

    int mrow = lane & 15;
    int k0   = (lane >> 4) << 1;                // 0 or 2
    int n    = lane & 15;
    v2f a, bm;
    a.x = wet[((b * 4 + k0)     << 12) + s0 + mrow];
    a.y = wet[((b * 4 + k0 + 1) << 12) + s0 + mrow];
    float bx = 0.f, by = 0.f, bias = 0.f;
    if (n < 12) { bx = in_w[n * 4 + k0]; by = in_w[n * 4 + k0 + 1]; bias = in_b[n]; }
    bm.x = bx; bm.y = by;
    v8f c;
    #pragma unroll
    for (int r = 0; r < 8; ++r) c[r] = bias;
    v8f d = __builtin_amdgcn_wmma_f32_16x16x4_f32(false, a, false, bm, (short)0, c, false, false);
    int mBase = (lane >> 4) << 3;
    if (n < 12) {
        #pragma unroll
        for (int r = 0; r < 8; ++r)
            qkv[(m0 + mBase + r) * 12 + n] = d[r];
    }
}

// ---------------------------------------------------------------- attention, head_dim = 1
// scores row i = q_i * k[:]  ->  row max = q_i * (q_i>=0 ? kmax : kmin)  (exact, no 2nd pass)
// k/v staged into LDS via CDNA5 async global->LDS copies; inner loop reads LDS as float4.
__global__ void attn_kernel(const float* __restrict__ qkv, float* __restrict__ attn_o) {
    __shared__ __align__(16) float k_s[4096];
    __shared__ __align__(16) float v_s[4096];
    __shared__ float rmax[256];
    __shared__ float rmin[256];
    int t = threadIdx.x;
    int bh = blockIdx.x >> 4;                   // 8 (b,h) pairs
    int split = blockIdx.x & 15;                // 16-way row split for occupancy
    int b = bh >> 2;
    int h = bh & 3;

    int i = split * 256 + t;                    // this thread's output row
    __builtin_prefetch(qkv + (size_t)(b * 4096 + i) * 12 + h, 0, 0); // -> global_prefetch_b8

    // async-stage k,v (32 ASYNCcnt slots per wave, counter is 6 bits)
    for (int j = t; j < 4096; j += 256) {
        const float* base = qkv + (size_t)(b * 4096 + j) * 12;
        async_load_f32_to_lds(base + 4 + h, &k_s[j]);
        async_load_f32_to_lds(base + 8 + h, &v_s[j]);
    }
    wait_asynccnt0();          // each wave waits for its own async copies
    __syncthreads();           // then make all waves' LDS writes visible

    // kmax/kmin over LDS
    float lmax = -3.4e38f, lmin = 3.4e38f;
    for (int j = t; j < 4096; j += 256) {
        float kv = k_s[j];
        lmax = fmaxf(lmax, kv); lmin = fminf(lmin, kv);
    }
    rmax[t] = lmax; rmin[t] = lmin;
    __syncthreads();
    for (int off = 128; off > 0; off >>= 1) {
        if (t < off) {
            rmax[t] = fmaxf(rmax[t], rmax[t + off]);
            rmin[t] = fminf(rmin[t], rmin[t + off]);
        }
        __syncthreads();
    }
    float kmax = rmax[0], kmin = rmin[0];

    float q = qkv[(size_t)(b * 4096 + i) * 12 + h];
    float m = (q >= 0.0f) ? q * kmax : q * kmin;
    float sum = 0.0f, acc = 0.0f;
    const float4* k4 = (const float4*)k_s;
    const float4* v4 = (const float4*)v_s;
    for (int j = 0; j < 1024; ++j) {            // ds_load_b128 instead of 4x b32
        float4 kk = k4[j];
        float4 vv = v4[j];
        float e0 = __expf(q * kk.x - m); sum += e0; acc += e0 * vv.x;
        float e1 = __expf(q * kk.y - m); sum += e1; acc += e1 * vv.y;
        float e2 = __expf(q * kk.z - m); sum += e2; acc += e2 * vv.z;
        float e3 = __expf(q * kk.w - m); sum += e3; acc += e3 * vv.w;
    }
    attn_o[(size_t)(b * 4096 + i) * 4 + h] = acc / sum;
}

// ---------------------------------------------------------------- attention out-proj via WMMA (K=4,N=4)
__global__ void out_proj_wmma_kernel(const float* __restrict__ attn_o,
                                     const float* __restrict__ ow, const float* __restrict__ ob,
                                     float* __restrict__ wetA) {
    int lane = threadIdx.x & 31;
    int wid  = threadIdx.x >> 5;
    int tile = blockIdx.x * 8 + wid;
    int m0 = tile << 4;
    int b  = m0 >> 12;
    int s0 = m0 & 4095;
    int mrow = lane & 15;
    int k0   = (lane >> 4) << 1;
    int n    = lane & 15;
    v2f a, bm;
    a.x = attn_o[(m0 + mrow) * 4 + k0];
    a.y = attn_o[(m0 + mrow) * 4 + k0 + 1];
    float bx = 0.f, by = 0.f, bias = 0.f;
    if (n < 4) { bx = ow[n * 4 + k0]; by = ow[n * 4 + k0 + 1]; bias = ob[n]; }
    bm.x = bx; bm.y = by;
    v8f c;
    #pragma unroll
    for (int r = 0; r < 8; ++r) c[r] = bias;
    v8f d = __builtin_amdgcn_wmma_f32_16x16x4_f32(false, a, false, bm, (short)0, c, false, false);
    int mBase = (lane >> 4) << 3;
    if (n < 4) {
        #pragma unroll
        for (int r = 0; r < 8; ++r)
            wetA[((b * 4 + n) << 12) + s0 + mBase + r] = d[r];    // back to [B,C,H,W]
    }
}

// ---------------------------------------------------------------- depthwise edge enhance
// wetB = wetA + (8*ctr - sum(neighbors)) * FEEDBACK*0.1
__global__ void edge_kernel(const float* __restrict__ wetA, float* __restrict__ wetB) {
    int tid = blockIdx.x * blockDim.x + threadIdx.x;   // 32768
    int p  = tid & 4095;
    int bc = tid >> 12;
    int y = p >> 6, x0 = p & 63;
    float ctr = wetA[tid];
    float acc = 0.0f;
    #pragma unroll
    for (int ky = -1; ky <= 1; ++ky) {
        int yy = y + ky;
        if ((unsigned)yy >= 64u) continue;
        #pragma unroll
        for (int kx = -1; kx <= 1; ++kx) {
            if (ky == 0 && kx == 0) continue;
            int xx = x0 + kx;
            if ((unsigned)xx >= 64u) continue;
            acc += wetA[(bc << 12) + (yy << 6) + xx];
        }
    }
    wetB[tid] = ctr + (8.0f * ctr - acc) * 0.04f;
}

// ---------------------------------------------------------------- final 1x1 (12->4) + dry mix, K split into 3 WMMAs
__global__ void final_wmma_kernel(const float* __restrict__ x,
                                  const float* __restrict__ spf, const float* __restrict__ wetB,
                                  const float* __restrict__ ow, const float* __restrict__ ob,
                                  float* __restrict__ out) {
    int lane = threadIdx.x & 31;
    int wid  = threadIdx.x >> 5;
    int tile = blockIdx.x * 8 + wid;
    int m0 = tile << 4;
    int b  = m0 >> 12;
    int s0 = m0 & 4095;
    int mrow = lane & 15;
    int k0   = (lane >> 4) << 1;
    int n    = lane & 15;
    float bias = (n < 4) ? ob[n] : 0.0f;
    v8f c;
    #pragma unroll
    for (int r = 0; r < 8; ++r) c[r] = bias;

    v2f a, bm;
    // K chunk 0: spatial channels 0..3
    a.x = spf[((b * 8 + k0)     << 12) + s0 + mrow];
    a.y = spf[((b * 8 + k0 + 1) << 12) + s0 + mrow];
    bm.x = (n < 4) ? ow[n * 12 + k0]     : 0.0f;
    bm.y = (n < 4) ? ow[n * 12 + k0 + 1] : 0.0f;
    c = __builtin_amdgcn_wmma_f32_16x16x4_f32(false, a, false, bm, (short)0, c, false, false);
    // K chunk 1: spatial channels 4..7
    a.x = spf[((b * 8 + 4 + k0)     << 12) + s0 + mrow];
    a.y = spf[((b * 8 + 4 + k0 + 1) << 12) + s0 + mrow];
    bm.x = (n < 4) ? ow[n * 12 + 4 + k0]     : 0.0f;
    bm.y = (n < 4) ? ow[n * 12 + 4 + k0 + 1] : 0.0f;
    c = __builtin_amdgcn_wmma_f32_16x16x4_f32(false, a, false, bm, (short)0, c, false, false);
    // K chunk 2: wet channels 0..3
    a.x = wetB[((b * 4 + k0)     << 12) + s0 + mrow];
    a.y = wetB[((b * 4 + k0 + 1) << 12) + s0 + mrow];
    bm.x = (n < 4) ? ow[n * 12 + 8 + k0]     : 0.0f;
    bm.y = (n < 4) ? ow[n * 12 + 8 + k0 + 1] : 0.0f;
    c = __builtin_amdgcn_wmma_f32_16x16x4_f32(false, a, false, bm, (short)0, c, false, false);

    int mBase = (lane >> 4) << 3;
    if (n < 4) {
        #pragma unroll
        for (int r = 0; r < 8; ++r) {
            int idx = ((b * 4 + n) << 12) + s0 + mBase + r;
            out[idx] = x[idx] * 0.7f + c[r] * 0.3f;     // dry*(1-WET_MIX) + processed*WET_MIX
        }
    }
}

// ---------------------------------------------------------------- host
extern "C" void kernel_launch(void* const* d_in, const int* in_sizes, int n_in,
                              void* d_out, int out_size, void* d_ws, size_t ws_size,
                              hipStream_t stream) {
    (void)in_sizes; (void)n_in; (void)out_size; (void)ws_size;
    const float* x    = (const float*)d_in[0];
    const float* rw   = (const float*)d_in[1];
    const float* rd   = (const float*)d_in[2];
    const float* sp_w = (const float*)d_in[3];
    const float* sp_b = (const float*)d_in[4];
    const float* fb_w = (const float*)d_in[5];
    const float* fb_b = (const float*)d_in[6];
    const float* ow   = (const float*)d_in[7];
    const float* ob   = (const float*)d_in[8];
    const float* d1_w = (const float*)d_in[9];
    const float* d1_b = (const float*)d_in[10];
    const float* d2_w = (const float*)d_in[11];
    const float* d2_b = (const float*)d_in[12];
    const float* ai_w = (const float*)d_in[13];
    const float* ai_b = (const float*)d_in[14];
    const float* ao_w = (const float*)d_in[15];
    const float* ao_b = (const float*)d_in[16];
    float* out = (float*)d_out;

    float* ws   = (float*)d_ws;       // ~1.4 MB total
    float* spf  = ws;                 // 65536  [B,8,HW]
    float* fb0  = ws + 65536;         // 32768
    float* fb1  = ws + 98304;         // 32768
    float* wet  = ws + 131072;        // 32768  [B,C,HW]
    float* qkv  = ws + 163840;        // 98304  [B,S,12]
    float* atno = ws + 262144;        // 32768  [B,S,4]
    float* wetA = ws + 294912;        // 32768
    float* wetB = ws + 327680;        // 32768

    init_kernel<<<128, 256, 0, stream>>>(x, fb0, wet);
    spatial_conv_kernel<<<256, 256, 0, stream>>>(x, sp_w, sp_b, spf);
    float* fbs[2] = { fb0, fb1 };
    float decay = 1.0f;                                  // 0.8^i
    for (int i = 0; i < 8; ++i) {
        reflection_step_kernel<<<32, 256, 0, stream>>>(fbs[i & 1], fbs[(i + 1) & 1], wet,
                                                       fb_w, fb_b, d1_w, d1_b, d2_w, d2_b,
                                                       rw, rd, i, decay);
        decay *= 0.8f;
    }
    qkv_proj_wmma_kernel<<<64, 256, 0, stream>>>(wet, ai_w, ai_b, qkv);
    attn_kernel<<<128, 256, 0, stream>>>(qkv, atno);
    out_proj_wmma_kernel<<<64, 256, 0, stream>>>(atno, ao_w, ao_b, wetA);
    edge_kernel<<<128, 256, 0, stream>>>(wetA, wetB);
    final_wmma_kernel<<<64, 256, 0, stream>>>(x, spf, wetB, ow, ob, out);
}